// VE_Loss_50508815401532
// MI455X (gfx1250) — compile-verified
//
#include <hip/hip_runtime.h>
#include <hip/hip_bf16.h>
#include <stdint.h>

// ---------------------------------------------------------------------------
// VE loss for pred[4,3,256,256] f32, lab[4,256,256] i32 -> scalar f32.
// Pipeline: argmax/compact -> row EDT pass (async->LDS staging) ->
//           col EDT pass (+plane max) -> cosine-sim + WMMA-ones reduction ->
//           finalize. All intermediates (~13 MB) stay L2-resident.
// ---------------------------------------------------------------------------

#define B_  4
#define C_  3
#define H_  256
#define W_  256
#define HW_ 65536            // H_*W_
#define NPLANE 24            // 12 lab planes + 12 pred planes
#define BIGF 1.0e9f
#define LIMITF 331776.0f     // 576*576
#define EPSF 1.0e-5f

typedef __attribute__((ext_vector_type(2))) float v2f;
typedef __attribute__((ext_vector_type(8))) float v8f;

// ------------------------------- K0: init ----------------------------------
__global__ void k_init(unsigned* plane_max, float* batch_sum, int* fg) {
    int t = threadIdx.x;
    if (t < NPLANE) plane_max[t] = 0u;
    if (t < B_) { batch_sum[t] = 0.0f; fg[t] = 0; }
}

// ----------------------- K1: argmax + compact labels -----------------------
__global__ void k_argmax(const float* __restrict__ pred,
                         const int*   __restrict__ lab,
                         unsigned char* __restrict__ predl,
                         unsigned char* __restrict__ labl,
                         int* __restrict__ fg) {
    int p = blockIdx.x * blockDim.x + threadIdx.x;       // 0 .. B*HW-1
    int b  = p >> 16;
    int hw = p & (HW_ - 1);
    const float* pp = pred + (size_t)b * C_ * HW_ + hw;
    float p0 = pp[0], p1 = pp[HW_], p2 = pp[2 * HW_];
    int arg = 0; float best = p0;
    if (p1 > best) { best = p1; arg = 1; }               // first-max tie rule
    if (p2 > best) { best = p2; arg = 2; }
    predl[p] = (unsigned char)arg;
    int l = lab[p];
    labl[p] = (unsigned char)l;
    // per-batch foreground flag; b is wave-uniform (wave spans 32 consecutive p)
    if (__any(l > 0) && ((threadIdx.x & 31) == 0)) atomicOr(&fg[b], 1);
}

// ------------------- K2: row pass of squared EDT (min-plus) ----------------
// grid (24, 32), block 256 = 8 waves; wave w handles row rowblk*8+w.
__global__ void k_rowpass(const unsigned char* __restrict__ labl,
                          const unsigned char* __restrict__ predl,
                          float* __restrict__ d1) {
    int plane = blockIdx.x;              // 0..23
    int wave  = threadIdx.x >> 5;        // 0..7
    int lane  = threadIdx.x & 31;
    int row   = blockIdx.y * 8 + wave;
    int b = (plane % 12) / C_;
    int c = plane % C_;
    const unsigned char* src =
        (plane < 12 ? labl : predl) + (size_t)b * HW_ + (size_t)row * W_;

    __shared__ unsigned char s_lbl[8][W_];

    // CDNA5 async copy: each lane moves 8 bytes global->LDS (ASYNCcnt path).
    unsigned ldsoff = (unsigned)(uintptr_t)(&s_lbl[wave][lane * 8]);
    unsigned goff   = (unsigned)(lane * 8);
    asm volatile("global_load_async_to_lds_b64 %0, %1, %2"
                 :
                 : "v"(ldsoff), "v"(goff), "s"(src)
                 : "memory");
    asm volatile("s_wait_asynccnt 0" ::: "memory");

    float d[8];
#pragma unroll
    for (int t = 0; t < 8; ++t) d[t] = 3.4e38f;
    for (int k = 0; k < W_; ++k) {
        float fk = (s_lbl[wave][k] == (unsigned char)c) ? BIGF : 0.0f;
#pragma unroll
        for (int t = 0; t < 8; ++t) {
            int j = lane + 32 * t;
            float df = (float)(j - k);
            d[t] = fminf(d[t], fmaf(df, df, fk));
        }
    }
    float* out = d1 + (size_t)plane * HW_ + (size_t)row * W_;
#pragma unroll
    for (int t = 0; t < 8; ++t) out[lane + 32 * t] = d[t];
}

// -------- K3: column pass + sqrt + per-plane max. grid (24,8), block 256 ----
__global__ void k_colpass(const float* __restrict__ d1,
                          float* __restrict__ dist,
                          unsigned* __restrict__ plane_max) {
    int plane = blockIdx.x;
    int j0    = blockIdx.y * 32;
    __shared__ float tile[H_][32];       // 32 KB of the 320 KB WGP LDS
    const float* src = d1 + (size_t)plane * HW_;
#pragma unroll 4
    for (int it = 0; it < 32; ++it) {
        int idx = it * 256 + threadIdx.x;      // 0..8191
        int k = idx >> 5, jj = idx & 31;
        tile[k][jj] = src[k * W_ + j0 + jj];
    }
    __syncthreads();

    int jj   = threadIdx.x & 31;
    int igrp = threadIdx.x >> 5;               // 0..7
    float vmax = 0.0f;
    float* outp = dist + (size_t)plane * HW_;
    for (int s = 0; s < 32; ++s) {
        int i = igrp * 32 + s;
        float dmin = 3.4e38f;
#pragma unroll 4
        for (int k = 0; k < H_; ++k) {
            float df = (float)(i - k);
            dmin = fminf(dmin, fmaf(df, df, tile[k][jj]));
        }
        float dd = sqrtf(dmin);
        outp[i * W_ + j0 + jj] = dd;
        vmax = fmaxf(vmax, dd);
    }
    for (int off = 16; off; off >>= 1)
        vmax = fmaxf(vmax, __shfl_xor(vmax, off, 32));
    if ((threadIdx.x & 31) == 0)
        atomicMax(&plane_max[plane], __float_as_uint(vmax));  // >=0 floats
}

// ------------------------- K4: cosine sim + WMMA reduce --------------------
// na = sqrt(sp^2+1) >= 1 and nb = sqrt(sl^2+1) >= 1, so the max(.,1e-8)
// guards in the reference are dead and
//   sim = dot/(na*nb) = dot * rsqrt((sp^2+1)*(sl^2+1))
// -> one native v_rsq_f32 instead of two sqrt-refine + two IEEE div chains.
__device__ __forceinline__ float simpix(int hw,
                                        const float* __restrict__ dl,
                                        const float* __restrict__ dp,
                                        const unsigned char* __restrict__ ll,
                                        const unsigned char* __restrict__ pl,
                                        const float* invL, const float* invP) {
    float sl = dl[hw] * invL[0] + dl[HW_ + hw] * invL[1] + dl[2 * HW_ + hw] * invL[2];
    float sp = dp[hw] * invP[0] + dp[HW_ + hw] * invP[1] + dp[2 * HW_ + hw] * invP[2];
    float dot = sp * sl + ((pl[hw] == ll[hw]) ? 1.0f : 0.0f);
    float n2  = fmaf(sp, sp, 1.0f) * fmaf(sl, sl, 1.0f);
    return dot * rsqrtf(n2);
}

// grid 64 blocks x 256 threads = 512 waves; each wave owns exactly 512 pixels
// of one batch (no tails, EXEC all-ones at every WMMA). Wave-local
// accumulation runs on the matrix unit: D = A(16x4 lane values) x B(ones) + C,
// so the f32 C tile is an exact 64-way adder per issue; sum(C) = 16 * total.
__global__ void k_loss(const unsigned char* __restrict__ labl,
                       const unsigned char* __restrict__ predl,
                       const float* __restrict__ dist,
                       const unsigned* __restrict__ plane_max,
                       float* __restrict__ batch_sum) {
    int gwave = (blockIdx.x * blockDim.x + threadIdx.x) >> 5;   // 0..511
    int lane  = threadIdx.x & 31;
    int base  = gwave * 512;
    int b     = base >> 16;
    int hwbase = base & (HW_ - 1);

    float invL[3], invP[3];
#pragma unroll
    for (int c = 0; c < C_; ++c) {
        float mL = __uint_as_float(plane_max[b * C_ + c]);
        float mP = __uint_as_float(plane_max[12 + b * C_ + c]);
        invL[c] = 1.0f / (fminf(mL, LIMITF) + EPSF);
        invP[c] = 1.0f / (fminf(mP, LIMITF) + EPSF);
    }
    const float* dl = dist + (size_t)(b * C_) * HW_;
    const float* dp = dist + (size_t)(12 + b * C_) * HW_;
    const unsigned char* ll = labl + (size_t)b * HW_;
    const unsigned char* pl = predl + (size_t)b * HW_;

    v8f acc = {};
    v2f ones; ones.x = 1.0f; ones.y = 1.0f;
#pragma unroll
    for (int it = 0; it < 8; ++it) {
        int h0 = hwbase + it * 64 + lane;
        v2f a;
        a.x = simpix(h0,      dl, dp, ll, pl, invL, invP);
        a.y = simpix(h0 + 32, dl, dp, ll, pl, invL, invP);
        acc = __builtin_amdgcn_wmma_f32_16x16x4_f32(
            /*neg_a=*/false, a, /*neg_b=*/false, ones,
            /*c_mod=*/(short)0, acc, /*reuse_a=*/false, /*reuse_b=*/false);
    }
    float ls = acc[0] + acc[1] + acc[2] + acc[3] +
               acc[4] + acc[5] + acc[6] + acc[7];
    for (int off = 16; off; off >>= 1) ls += __shfl_xor(ls, off, 32);
    if (lane == 0) atomicAdd(&batch_sum[b], ls * (1.0f / 16.0f));
}

// ------------------------------ K5: finalize -------------------------------
__global__ void k_final(const float* __restrict__ batch_sum,
                        const int* __restrict__ fg,
                        float* __restrict__ out) {
    int lane = threadIdx.x;
    float v = 0.0f;
    if (lane < B_) {
        float lb = 1.0f - batch_sum[lane] * (1.0f / (float)HW_);
        v = fg[lane] ? lb : 0.0f;
    }
    for (int off = 16; off; off >>= 1) v += __shfl_xor(v, off, 32);
    if (lane == 0) {
        float loss = v * (1.0f / (float)B_);
        if (isnan(loss)) loss = 0.0f;
        else if (isinf(loss)) loss = (loss > 0.0f) ? 1.0f : 0.0f;
        out[0] = loss;
    }
}

// ---------------------------------------------------------------------------
extern "C" void kernel_launch(void* const* d_in, const int* in_sizes, int n_in,
                              void* d_out, int out_size, void* d_ws, size_t ws_size,
                              hipStream_t stream) {
    const float* pred = (const float*)d_in[0];   // [4,3,256,256] f32
    const int*   lab  = (const int*)d_in[1];     // [4,256,256] i32
    float* out = (float*)d_out;

    char* ws = (char*)d_ws;
    float*         d1        = (float*)(ws);                             // 24*HW f32
    float*         dist      = (float*)(ws + 6291456);                   // 24*HW f32
    unsigned char* predl     = (unsigned char*)(ws + 12582912);          // B*HW u8
    unsigned char* labl      = (unsigned char*)(ws + 12845056);          // B*HW u8
    unsigned*      plane_max = (unsigned*)(ws + 13107200);               // 24 u32
    float*         batch_sum = (float*)(ws + 13107296);                  // 4 f32
    int*           fg        = (int*)(ws + 13107312);                    // 4 i32

    k_init<<<1, 64, 0, stream>>>(plane_max, batch_sum, fg);
    k_argmax<<<(B_ * HW_) / 256, 256, 0, stream>>>(pred, lab, predl, labl, fg);
    k_rowpass<<<dim3(NPLANE, 32), 256, 0, stream>>>(labl, predl, d1);
    k_colpass<<<dim3(NPLANE, 8), 256, 0, stream>>>(d1, dist, plane_max);
    k_loss<<<64, 256, 0, stream>>>(labl, predl, dist, plane_max, batch_sum);
    k_final<<<1, 32, 0, stream>>>(batch_sum, fg, out);
}